// FederatedImbalanceLoss_v2_78108275245519
// MI455X (gfx1250) — compile-verified
//
#include <hip/hip_runtime.h>
#include <math.h>

// Problem constants (match reference)
#define BATCH 4096
#define NCLS  256
#define DIM   64
#define TAO_F 1.0f
#define EPS_F 1e-6f

typedef __attribute__((ext_vector_type(2))) float v2f;
typedef __attribute__((ext_vector_type(8))) float v8f;

// Workspace layout (floats):
//   ws[0..255]   : per-class cos2 numerator-sum accumulator (atomic)
//   ws[256]      : loss accumulator (atomic)
//   ws[257..512] : proto_factor per class

__global__ void k_init(float* __restrict__ ws) {
    int i = blockIdx.x * blockDim.x + threadIdx.x;
    if (i < NCLS + 1) ws[i] = 0.0f;
}

// Streams batch_protos [B,C,D] once (268 MB > 192 MB L2 -> non-temporal).
// One wave handles 16 b's at fixed class c.
// num[b] = dot(bp[b,c,:], gp[c,:]) via v_wmma_f32_16x16x4_f32 accumulation,
// sq[b]  = ||bp[b,c,:]||^2 via per-lane packed FMAs + shfl_xor(16).
__global__ __launch_bounds__(256) void k_cos2(const float* __restrict__ bp,
                                              const float* __restrict__ gp,
                                              float* __restrict__ cos2sum) {
    const int lane = threadIdx.x & 31;
    const int wave = threadIdx.x >> 5;
    const int c    = blockIdx.y;
    const int b0   = (blockIdx.x * 8 + wave) * 16;
    const int m    = lane & 15;   // row within 16-row tile (A matrix M index)
    const int h    = lane >> 4;   // K half-group: this lane holds K = 2h, 2h+1

    const float* __restrict__ rowA = bp + ((size_t)(b0 + m) * NCLS + c) * DIM + 2 * h;
    const float* __restrict__ rowG = gp + (size_t)c * DIM + 2 * h;

    v8f   acc    = {};     // D/C accumulator: num[m] replicated across columns
    float sqpart = 0.0f;   // partial ||bp row||^2 (this lane's K half)
    float gpart  = 0.0f;   // partial ||gp[c]||^2  (this lane's K half)

    #pragma unroll
    for (int kk = 0; kk < 16; ++kk) {         // 16 chunks of K=4 cover D=64
        // Read-once stream: non-temporal so it doesn't thrash L2 (268 MB total)
        v2f a = __builtin_nontemporal_load(
                    reinterpret_cast<const v2f*>(rowA + kk * 4));
        // gp[c] is reused by all b-tiles of class c: keep it cached
        v2f g = *reinterpret_cast<const v2f*>(rowG + kk * 4);
        sqpart = fmaf(a.x, a.x, fmaf(a.y, a.y, sqpart));
        gpart  = fmaf(g.x, g.x, fmaf(g.y, g.y, gpart));
        // A[m, 2h + {0,1}] ; B[k, n] = gp[k] broadcast across all 16 columns
        acc = __builtin_amdgcn_wmma_f32_16x16x4_f32(
            /*neg_a=*/false, a, /*neg_b=*/false, g,
            /*c_mod=*/(short)0, acc, /*reuse_a=*/false, /*reuse_b=*/false);
    }

    // Combine K halves across the two 16-lane groups
    float sq  = sqpart + __shfl_xor(sqpart, 16, 32);   // ||bp[b0+m,c,:]||^2
    float gn2 = gpart  + __shfl_xor(gpart, 16, 32);    // ||gp[c,:]||^2
    float w   = 1.0f / (sqrtf(sq) * sqrtf(gn2) + EPS_F);

    // Diagonal ownership: lane holds rows M = v + 8h in acc[v]; own row m iff
    // v_own = m - 8h in [0,8)  -> lanes 0..7 (m=0..7) and 24..31 (m=8..15).
    const int v_own = m - 8 * h;
    float contrib = 0.0f;
    #pragma unroll
    for (int v = 0; v < 8; ++v)
        contrib += (v_own == v) ? acc[v] * w : 0.0f;

    // Wave reduction, then one atomic per (c, 16-row tile)
    #pragma unroll
    for (int off = 16; off > 0; off >>= 1)
        contrib += __shfl_xor(contrib, off, 32);
    if (lane == 0) atomicAdd(&cos2sum[c], contrib);
}

// cos1 per class + proto_factor. Grid = NCLS blocks of 64 threads (d index).
__global__ void k_pf(const float* __restrict__ lp,
                     const float* __restrict__ gp,
                     float* __restrict__ ws) {
    __shared__ float s0[64], s1[64], s2[64];
    const int c = blockIdx.x, d = threadIdx.x;
    float a = lp[(size_t)c * DIM + d];
    float b = gp[(size_t)c * DIM + d];
    s0[d] = a * b; s1[d] = a * a; s2[d] = b * b;
    __syncthreads();
    #pragma unroll
    for (int off = 32; off > 0; off >>= 1) {
        if (d < off) { s0[d] += s0[d + off]; s1[d] += s1[d + off]; s2[d] += s2[d + off]; }
        __syncthreads();
    }
    if (d == 0) {
        float cos1 = s0[0] / (sqrtf(s1[0]) * sqrtf(s2[0]) + EPS_F);
        float cos2 = ws[c] * (1.0f / (float)BATCH);
        ws[257 + c] = (1.0f + TAO_F) / ((cos1 + cos2) * 0.5f + TAO_F);
    }
}

// Per-row loss: denom[b] = sum_j gf[tgt,j] * exp(logit - rowmax).
// gf from tier structure: gf = 2^(2.4*(ti - tj)) when ti < tj else 1.
__global__ __launch_bounds__(256) void k_loss(const float* __restrict__ logits,
                                              const int* __restrict__ targets,
                                              const float* __restrict__ ws,
                                              float* __restrict__ lossacc) {
    __shared__ float red[256];
    __shared__ float s_etgt;
    const int b = blockIdx.x, j = threadIdx.x;
    const int tgt = targets[b];
    float x = __builtin_nontemporal_load(logits + (size_t)b * NCLS + j);

    red[j] = x;
    __syncthreads();
    #pragma unroll
    for (int off = 128; off > 0; off >>= 1) {
        if (j < off) red[j] = fmaxf(red[j], red[j + off]);
        __syncthreads();
    }
    float mx = red[0];
    __syncthreads();

    float e = expf(x - mx);
    if (j == tgt) s_etgt = e;
    int ti = tgt >> 6, tj = j >> 6;
    float gf = (ti < tj) ? exp2f(2.4f * (float)(ti - tj)) : 1.0f;
    red[j] = gf * e;
    __syncthreads();
    #pragma unroll
    for (int off = 128; off > 0; off >>= 1) {
        if (j < off) red[j] += red[j + off];
        __syncthreads();
    }
    if (j == 0) {
        float denom = red[0];
        float sigma = s_etgt / (denom + EPS_F);
        float pf = ws[257 + tgt];
        atomicAdd(lossacc, -pf * logf(sigma + EPS_F));
    }
}

__global__ void k_final(const float* __restrict__ lossacc, float* __restrict__ out) {
    out[0] = lossacc[0] * (1.0f / (float)BATCH);
}

extern "C" void kernel_launch(void* const* d_in, const int* in_sizes, int n_in,
                              void* d_out, int out_size, void* d_ws, size_t ws_size,
                              hipStream_t stream) {
    (void)in_sizes; (void)n_in; (void)out_size; (void)ws_size;
    const float* logits       = (const float*)d_in[0];
    const int*   targets      = (const int*)  d_in[1];
    const float* local_proto  = (const float*)d_in[2];
    const float* global_proto = (const float*)d_in[3];
    const float* batch_protos = (const float*)d_in[4];
    float* out = (float*)d_out;
    float* ws  = (float*)d_ws;

    k_init<<<2, 256, 0, stream>>>(ws);
    // 32 x-blocks * 8 waves * 16 rows = 4096 = B exactly; y = class.
    k_cos2<<<dim3(32, NCLS), 256, 0, stream>>>(batch_protos, global_proto, ws);
    k_pf<<<NCLS, 64, 0, stream>>>(local_proto, global_proto, ws);
    k_loss<<<BATCH, 256, 0, stream>>>(logits, targets, ws, ws + 256);
    k_final<<<1, 1, 0, stream>>>(ws + 256, out);
}